// GhostVLAD_26663156974220
// MI455X (gfx1250) — compile-verified
//
#include <hip/hip_runtime.h>
#include <math.h>

typedef __attribute__((ext_vector_type(2))) float v2f;
typedef __attribute__((ext_vector_type(8))) float v8f;

#define C_DIM  512
#define P_DIM  1024
#define KPAD   16
#define KG     10
#define VLADK  8
#define NBATCH 32
#define PCHUNK 8          // pixel chunks per image for the aggregation GEMM
#define PCL    (P_DIM / PCHUNK)   // 128 pixels per chunk

// ---------------------------------------------------------------------------
// Kernel A: logits = conv_w . x + b  ->  softmax over k  ->  a  (to d_ws)
// Grid: (P/128, N), block 256 (8 waves). Each wave owns a 16-pixel tile.
// WMMA: D(16k x 16p) += A(16k x 4c) * B(4c x 16p), K = C in steps of 4.
// ---------------------------------------------------------------------------
__global__ __launch_bounds__(256)
void ghostvlad_assign(const float* __restrict__ x,
                      const float* __restrict__ conv_w,
                      const float* __restrict__ conv_b,
                      float* __restrict__ aBuf) {
  // W staged as [c/2][row][2] so the A-fragment is one conflict-free b64 ds load
  __shared__ float wLds[C_DIM * KPAD];   // 32 KB
  __shared__ float sBias[KPAD];

  const int tid  = threadIdx.x;
  const int lane = tid & 31;
  const int wave = tid >> 5;
  const int n    = blockIdx.y;
  const int p0   = blockIdx.x * 128 + wave * 16;

  for (int idx = tid; idx < C_DIM * KPAD; idx += 256) {
    const int cp   = idx >> 5;
    const int rem  = idx & 31;
    const int row  = rem >> 1;
    const int half = rem & 1;
    const int c    = cp * 2 + half;
    wLds[idx] = (row < KG) ? conv_w[row * C_DIM + c] : 0.0f;
  }
  if (tid < KPAD) sBias[tid] = (tid < KG) ? conv_b[tid] : 0.0f;
  __syncthreads();

  const int  row = lane & 15;           // A-row (k) == B-col (pixel) index
  const int  kb  = (lane >> 4) << 1;    // K sub-offset {0,2}
  const bool hi  = (lane >= 16);
  const float* xn = x + (size_t)n * C_DIM * P_DIM;

  v8f acc = {};
  for (int kc = 0; kc < C_DIM; kc += 4) {
    const int cpair = (kc + kb) >> 1;
    v2f afrag = *(const v2f*)&wLds[cpair * 32 + row * 2];
    const float* bp = xn + (size_t)(kc + kb) * P_DIM + p0 + row;
    v2f bfrag;
    bfrag[0] = bp[0];
    bfrag[1] = bp[P_DIM];
    acc = __builtin_amdgcn_wmma_f32_16x16x4_f32(false, afrag, false, bfrag,
                                                (short)0, acc, false, false);
  }

  // C layout: VGPR j, lanes 0-15 -> M=j, lanes 16-31 -> M=j+8, N = lane&15.
  float cv[8];
#pragma unroll
  for (int j = 0; j < 8; ++j) cv[j] = acc[j];

  const int nvalid = hi ? 2 : 8;        // valid softmax rows: k = 0..9
  const int Mbase  = hi ? 8 : 0;

  float m = -3.4e38f;
#pragma unroll
  for (int j = 0; j < 8; ++j)
    if (j < nvalid) { cv[j] += sBias[Mbase + j]; m = fmaxf(m, cv[j]); }
  m = fmaxf(m, __shfl_xor(m, 16));

  float s = 0.0f;
#pragma unroll
  for (int j = 0; j < 8; ++j) {
    if (j < nvalid) { cv[j] = __expf(cv[j] - m); s += cv[j]; }
    else            { cv[j] = 0.0f; }
  }
  s += __shfl_xor(s, 16);
  const float inv = 1.0f / s;
#pragma unroll
  for (int j = 0; j < 8; ++j) cv[j] *= inv;

  // a[n][M][p], k padded to 16 with zeros
  float* ap = aBuf + ((size_t)n * KPAD + Mbase) * P_DIM + p0 + row;
#pragma unroll
  for (int j = 0; j < 8; ++j) ap[(size_t)j * P_DIM] = cv[j];
}

// ---------------------------------------------------------------------------
// Kernel B: partial ax[k][c] = sum_{p in chunk} a[k][p] x[c][p].
// Grid: (PCHUNK, N), block 256 (8 waves); wave w owns c-range [64w, 64w+64)
// as 4 WMMA c-tiles, reusing the A-fragment across all 4.
// WMMA: D(16k x 16c) += A(16k x 4p) * B(4p x 16c), K = 128-pixel chunk.
// x is re-read here but stays L2-resident (67 MB < 192 MB L2); 256 blocks
// keep enough WGPs busy to pull L2 at full rate.
// ---------------------------------------------------------------------------
__global__ __launch_bounds__(256)
void ghostvlad_aggregate(const float* __restrict__ x,
                         const float* __restrict__ aBuf,
                         float* __restrict__ part) {
  const int tid  = threadIdx.x;
  const int lane = tid & 31;
  const int wave = tid >> 5;
  const int pc   = blockIdx.x;
  const int n    = blockIdx.y;

  const int col = lane & 15;            // A-row (k) == B-col (c) index
  const int kb  = (lane >> 4) << 1;     // K sub-offset {0,2}
  const int c0  = wave * 64;
  const int pbase = pc * PCL;

  const float* xn = x    + (size_t)n * C_DIM * P_DIM;
  const float* an = aBuf + (size_t)n * KPAD * P_DIM;

  v8f acc0 = {}, acc1 = {}, acc2 = {}, acc3 = {};
  for (int kp = pbase; kp < pbase + PCL; kp += 4) {
    v2f afrag = *(const v2f*)(an + (size_t)col * P_DIM + kp + kb);
    v2f bf;
    bf = *(const v2f*)(xn + (size_t)(c0 +  0 + col) * P_DIM + kp + kb);
    acc0 = __builtin_amdgcn_wmma_f32_16x16x4_f32(false, afrag, false, bf,
                                                 (short)0, acc0, false, false);
    bf = *(const v2f*)(xn + (size_t)(c0 + 16 + col) * P_DIM + kp + kb);
    acc1 = __builtin_amdgcn_wmma_f32_16x16x4_f32(false, afrag, false, bf,
                                                 (short)0, acc1, false, false);
    bf = *(const v2f*)(xn + (size_t)(c0 + 32 + col) * P_DIM + kp + kb);
    acc2 = __builtin_amdgcn_wmma_f32_16x16x4_f32(false, afrag, false, bf,
                                                 (short)0, acc2, false, false);
    bf = *(const v2f*)(xn + (size_t)(c0 + 48 + col) * P_DIM + kp + kb);
    acc3 = __builtin_amdgcn_wmma_f32_16x16x4_f32(false, afrag, false, bf,
                                                 (short)0, acc3, false, false);
  }

  // Keep only k = 0..7 (VLAD clusters): lanes 0-15 hold M = j.
  // part layout: [n][pc][k(8)][c(512)]
  if (lane < 16) {
    float* pp = part + (((size_t)n * PCHUNK + pc) * VLADK) * C_DIM;
#pragma unroll
    for (int j = 0; j < 8; ++j) {
      pp[j * C_DIM + c0 +  0 + col] = acc0[j];
      pp[j * C_DIM + c0 + 16 + col] = acc1[j];
      pp[j * C_DIM + c0 + 32 + col] = acc2[j];
      pp[j * C_DIM + c0 + 48 + col] = acc3[j];
    }
  }
}

// ---------------------------------------------------------------------------
// Kernel C: finalize. Per (n,k): sum the 8 partial ax chunks, subtract
// a_sum[k]*centers[k], L2-normalize over c, write out. All reductions are
// fixed-order (deterministic, no atomics). Tiny: ~6 MB total traffic.
// Grid: N, block 256 (8 waves); wave k owns cluster k.
// ---------------------------------------------------------------------------
__global__ __launch_bounds__(256)
void ghostvlad_finalize(const float* __restrict__ centers,
                        const float* __restrict__ aBuf,
                        const float* __restrict__ part,
                        float* __restrict__ out) {
  const int tid  = threadIdx.x;
  const int lane = tid & 31;
  const int k    = tid >> 5;            // 8 waves -> k = 0..7
  const int n    = blockIdx.x;

  const float* an = aBuf + ((size_t)n * KPAD + k) * P_DIM;

  // a_sum[n][k]: fixed-order reduction over all 1024 pixels
  float s = 0.0f;
  for (int i = lane; i < P_DIM; i += 32) s += an[i];
  s += __shfl_xor(s, 1);  s += __shfl_xor(s, 2);
  s += __shfl_xor(s, 4);  s += __shfl_xor(s, 8);
  s += __shfl_xor(s, 16);

  const float* pp = part + ((size_t)n * PCHUNK * VLADK + k) * C_DIM;
  const float* ck = centers + (size_t)k * C_DIM;

  float v[16];
  float ss = 0.0f;
#pragma unroll
  for (int i = 0; i < 16; ++i) {
    const int c = lane + i * 32;
    float r = 0.0f;
#pragma unroll
    for (int pc = 0; pc < PCHUNK; ++pc)
      r += pp[(size_t)pc * VLADK * C_DIM + c];
    r -= s * ck[c];
    v[i] = r;
    ss += r * r;
  }
  ss += __shfl_xor(ss, 1);  ss += __shfl_xor(ss, 2);
  ss += __shfl_xor(ss, 4);  ss += __shfl_xor(ss, 8);
  ss += __shfl_xor(ss, 16);

  const float inv = 1.0f / fmaxf(sqrtf(ss), 1e-12f);
  float* op = out + (size_t)n * (VLADK * C_DIM) + (size_t)k * C_DIM;
#pragma unroll
  for (int i = 0; i < 16; ++i) op[lane + i * 32] = v[i] * inv;
}

// ---------------------------------------------------------------------------
extern "C" void kernel_launch(void* const* d_in, const int* in_sizes, int n_in,
                              void* d_out, int out_size, void* d_ws, size_t ws_size,
                              hipStream_t stream) {
  const float* x       = (const float*)d_in[0];   // (32, 512, 16, 64)
  const float* conv_w  = (const float*)d_in[1];   // (10, 512)
  const float* conv_b  = (const float*)d_in[2];   // (10,)
  const float* centers = (const float*)d_in[3];   // (10, 512)
  float*       out     = (float*)d_out;           // (32, 4096)

  float* aBuf = (float*)d_ws;                               // 2 MB
  float* part = aBuf + (size_t)NBATCH * KPAD * P_DIM;       // 4 MB

  dim3 gA(P_DIM / 128, NBATCH);
  ghostvlad_assign<<<gA, 256, 0, stream>>>(x, conv_w, conv_b, aBuf);

  dim3 gB(PCHUNK, NBATCH);
  ghostvlad_aggregate<<<gB, 256, 0, stream>>>(x, aBuf, part);

  ghostvlad_finalize<<<NBATCH, 256, 0, stream>>>(centers, aBuf, part, out);
}